// CompGCN_52493090292006
// MI455X (gfx1250) — compile-verified
//
#include <hip/hip_runtime.h>
#include <hip/hip_bf16.h>
#include <math.h>

// Problem constants (match reference)
#define DD    100     // embedding dim
#define OUTD  200     // output dim
#define NT16  13      // ceil(OUTD/16) N-tiles
#define NPAD  (NT16 * 16)   // 208: padded N so no guards in WMMA K-loops
#define ET    16      // edges per block in edge kernel

typedef __attribute__((ext_vector_type(2))) float v2f;
typedef __attribute__((ext_vector_type(8))) float v8f;

__device__ __forceinline__ v8f wmma_f32(v2f a, v2f b, v8f c) {
  return __builtin_amdgcn_wmma_f32_16x16x4_f32(
      /*neg_a=*/false, a, /*neg_b=*/false, b,
      /*c_mod=*/(short)0, c, /*reuse_a=*/false, /*reuse_b=*/false);
}

// ---------------------------------------------------------------------------
// Prep: build padded TRANSPOSED B-operands in workspace, [NPAD][DD] layout:
//   WTin[c][k]  = in_w[k][c]          (0 for c >= OUTD)
//   WTout[c][k] = out_w[k][c]
//   WTrel[c][k] = w_rel[k][c]
//   MT[c][j]    = sum_k loop_rel[(j+k)%D] * loop_w[k][c]   (ccorr bilinearity)
// Transposed layout makes each lane's B-frag K-pair contiguous -> b64 loads.
// ---------------------------------------------------------------------------
__global__ void prep_kernel(const float* __restrict__ in_w,
                            const float* __restrict__ out_w,
                            const float* __restrict__ w_rel,
                            const float* __restrict__ loop_rel,
                            const float* __restrict__ loop_w,
                            float* __restrict__ WTin,
                            float* __restrict__ WTout,
                            float* __restrict__ WTrel,
                            float* __restrict__ MT) {
  int i = blockIdx.x * blockDim.x + threadIdx.x;   // over NPAD*DD
  if (i >= NPAD * DD) return;
  int c = i / DD, k = i % DD;
  const bool ok = c < OUTD;
  WTin[i]  = ok ? in_w [k * OUTD + c] : 0.f;
  WTout[i] = ok ? out_w[k * OUTD + c] : 0.f;
  WTrel[i] = ok ? w_rel[k * OUTD + c] : 0.f;
  float s = 0.f;
  if (ok) {
    const int j = k;
    #pragma unroll 4
    for (int kk = 0; kk < DD; ++kk) {
      int idx = j + kk; if (idx >= DD) idx -= DD;
      s += loop_rel[idx] * loop_w[kk * OUTD + c];
    }
  }
  MT[i] = s;
}

// ---------------------------------------------------------------------------
// Self-loop GEMM via WMMA: agg = ent_emb @ M_loop / 3 + conv_bias  (store)
// M = V (3125 tiles), N = OUT (13 tiles), K = 100 (25 WMMA steps).
// ---------------------------------------------------------------------------
__global__ void loop_wmma_kernel(const float* __restrict__ ent_emb,   // [V, D]
                                 const float* __restrict__ MT,        // [NPAD, D]
                                 const float* __restrict__ conv_bias, // [OUT]
                                 float* __restrict__ agg) {           // [V, OUT]
  const int wave = threadIdx.x >> 5;
  const int lane = threadIdx.x & 31;
  const int nt = blockIdx.y * 8 + wave;
  if (nt >= NT16) return;                       // wave-uniform exit
  const int m0 = blockIdx.x * 16;
  const int row = lane & 15;
  const int kh  = (lane >> 4) << 1;             // 0 or 2
  const int col = nt * 16 + row;                // < NPAD always

  const float* __restrict__ aPtr = ent_emb + (size_t)(m0 + row) * DD + kh;
  const float* __restrict__ bPtr = MT      + (size_t)col * DD + kh;

  v8f c = {};
  for (int k = 0; k < DD; k += 4) {
    v2f a = *(const v2f*)(aPtr + k);
    v2f b = *(const v2f*)(bPtr + k);
    c = wmma_f32(a, b, c);
  }
  if (col < OUTD) {                             // guard only in epilogue
    const float bias = conv_bias[col];
    const int mOff = (lane >> 4) * 8;
    #pragma unroll
    for (int g = 0; g < 8; ++g)
      agg[(size_t)(m0 + g + mOff) * OUTD + col] = c[g] * (1.0f / 3.0f) + bias;
  }
}

// ---------------------------------------------------------------------------
// Fused edge kernel: per-edge ccorr in LDS (scalar, irregular operands),
// then 16-edge x 16-out projection tiles via WMMA (A from LDS, B = WT[col][k]
// contiguous b64 loads). Epilogue: atomic scatter-add msg*norm/3 into agg.
// Block = 16 edges, 256 threads (8 waves); waves loop over 13 N-tiles.
// ---------------------------------------------------------------------------
__global__ void edge_wmma_kernel(const int* __restrict__ edge_src,
                                 const int* __restrict__ edge_dst,
                                 const int* __restrict__ edge_type,
                                 const float* __restrict__ edge_norm,
                                 const float* __restrict__ ent_emb,
                                 const float* __restrict__ rel_emb,
                                 const float* __restrict__ WTin,   // [NPAD, D]
                                 const float* __restrict__ WTout,  // [NPAD, D]
                                 float* __restrict__ agg,
                                 int Ehalf) {
  __shared__ float sa[ET][DD];        // src embeddings
  __shared__ float sb[ET][DD];        // relation embeddings
  __shared__ float sc[ET][DD + 4];    // ccorr results (WMMA A-matrix), padded
  __shared__ int   sdst[ET];
  __shared__ float sscale[ET];

  const int e0  = blockIdx.x * ET;
  const int tid = threadIdx.x;

  for (int i = tid; i < ET * DD; i += 256) {
    int er = i / DD, k = i % DD;
    int e = e0 + er;
    sa[er][k] = ent_emb[(size_t)edge_src[e] * DD + k];
    sb[er][k] = rel_emb[(size_t)edge_type[e] * DD + k];
  }
  if (tid < ET) {
    int e = e0 + tid;
    sdst[tid]   = edge_dst[e];
    sscale[tid] = edge_norm[e] * (1.0f / 3.0f);
  }
  __syncthreads();

  // ccorr: c[er][k] = sum_j a[j] * b[(j+k)%D]   (1600 outputs / block)
  for (int i = tid; i < ET * DD; i += 256) {
    int er = i / DD, k = i % DD;
    const float* __restrict__ a = sa[er];
    const float* __restrict__ b = sb[er];
    float s = 0.f;
    #pragma unroll 4
    for (int j = 0; j < DD; ++j) {
      int idx = j + k; if (idx >= DD) idx -= DD;
      s += a[j] * b[idx];
    }
    sc[er][k] = s;
  }
  __syncthreads();

  // projection tiles: msg = ccorr @ W  (W block-uniform: Ehalf % ET == 0)
  const float* __restrict__ WT = (e0 < Ehalf) ? WTin : WTout;
  const int wave = tid >> 5;
  const int lane = tid & 31;
  const int row  = lane & 15;
  const int kh   = (lane >> 4) << 1;

  for (int nt = wave; nt < NT16; nt += 8) {
    const int col = nt * 16 + row;
    const float* __restrict__ bPtr = WT + (size_t)col * DD + kh;
    v8f acc = {};
    for (int k = 0; k < DD; k += 4) {
      v2f a = *(const v2f*)&sc[row][k + kh];    // ds_load_b64
      v2f b = *(const v2f*)(bPtr + k);          // global_load_b64
      acc = wmma_f32(a, b, acc);
    }
    if (col < OUTD) {
      const int mOff = (lane >> 4) * 8;
      #pragma unroll
      for (int g = 0; g < 8; ++g) {
        const int er = g + mOff;
        atomicAdd(&agg[(size_t)sdst[er] * OUTD + col], acc[g] * sscale[er]);
      }
    }
  }
}

// ---------------------------------------------------------------------------
// r_out = rel_emb @ w_rel via WMMA: M = R (25 tiles), N = 13 tiles, K = 100.
// ---------------------------------------------------------------------------
__global__ void relout_wmma_kernel(const float* __restrict__ rel_emb, // [R, D]
                                   const float* __restrict__ WTrel,   // [NPAD, D]
                                   float* __restrict__ r_out) {       // [R, OUT]
  const int wave = threadIdx.x >> 5;
  const int lane = threadIdx.x & 31;
  const int nt = blockIdx.y * 8 + wave;
  if (nt >= NT16) return;
  const int m0 = blockIdx.x * 16;
  const int row = lane & 15;
  const int kh  = (lane >> 4) << 1;
  const int col = nt * 16 + row;

  const float* __restrict__ aPtr = rel_emb + (size_t)(m0 + row) * DD + kh;
  const float* __restrict__ bPtr = WTrel   + (size_t)col * DD + kh;

  v8f c = {};
  for (int k = 0; k < DD; k += 4) {
    v2f a = *(const v2f*)(aPtr + k);
    v2f b = *(const v2f*)(bPtr + k);
    c = wmma_f32(a, b, c);
  }
  if (col < OUTD) {
    const int mOff = (lane >> 4) * 8;
    #pragma unroll
    for (int g = 0; g < 8; ++g)
      r_out[(size_t)(m0 + g + mOff) * OUTD + col] = c[g];
  }
}

// ---------------------------------------------------------------------------
// BatchNorm statistics: per-column sum / sumsq over V rows
// ---------------------------------------------------------------------------
__global__ void bn_stats_kernel(const float* __restrict__ x,
                                float* __restrict__ sum,
                                float* __restrict__ sumsq,
                                int n) {
  __shared__ float s[OUTD], ss[OUTD];
  for (int i = threadIdx.x; i < OUTD; i += blockDim.x) { s[i] = 0.f; ss[i] = 0.f; }
  __syncthreads();
  for (int i = blockIdx.x * blockDim.x + threadIdx.x; i < n;
       i += gridDim.x * blockDim.x) {
    float v = x[i];
    int o = i % OUTD;
    atomicAdd(&s[o], v);
    atomicAdd(&ss[o], v * v);
  }
  __syncthreads();
  for (int i = threadIdx.x; i < OUTD; i += blockDim.x) {
    atomicAdd(&sum[i], s[i]);
    atomicAdd(&sumsq[i], ss[i]);
  }
}

// ---------------------------------------------------------------------------
// BN apply + tanh (in place); population variance as in jnp.var
// ---------------------------------------------------------------------------
__global__ void bn_apply_kernel(float* __restrict__ x,
                                const float* __restrict__ sum,
                                const float* __restrict__ sumsq,
                                const float* __restrict__ gamma,
                                const float* __restrict__ beta,
                                int n, float invV) {
  int i = blockIdx.x * blockDim.x + threadIdx.x;
  if (i >= n) return;
  int o = i % OUTD;
  float m   = sum[o] * invV;
  float var = sumsq[o] * invV - m * m;
  float v = (x[i] - m) * rsqrtf(var + 1e-5f) * gamma[o] + beta[o];
  x[i] = tanhf(v);
}

// ---------------------------------------------------------------------------
// obj[b,:] = x[head[b],:] * r_out[rela[b],:]
// ---------------------------------------------------------------------------
__global__ void obj_kernel(const float* __restrict__ x,
                           const float* __restrict__ r_out,
                           const int* __restrict__ head,
                           const int* __restrict__ rela,
                           float* __restrict__ obj,
                           int n) {
  int i = blockIdx.x * blockDim.x + threadIdx.x;
  if (i >= n) return;
  int b = i / OUTD, o = i % OUTD;
  obj[i] = x[head[b] * OUTD + o] * r_out[rela[b] * OUTD + o];
}

// ---------------------------------------------------------------------------
// Decoder GEMM + sigmoid via WMMA: score = sigmoid(obj @ emb_ent^T + bias)
// M = B (64 tiles), N = V (3125 tiles), K = OUT = 200 (50 WMMA steps).
// Store-bandwidth bound (205 MB out); emb_ent (40 MB) is L2-resident.
// ---------------------------------------------------------------------------
__global__ void decoder_wmma_kernel(const float* __restrict__ obj,      // [B, K]
                                    const float* __restrict__ emb_ent,  // [V, K]
                                    const float* __restrict__ ent_bias, // [V]
                                    float* __restrict__ score,          // [B, V]
                                    int Vn, int Kn) {
  const int wave = threadIdx.x >> 5;
  const int lane = threadIdx.x & 31;
  const int nTile = blockIdx.x * 8 + wave;
  const int nTiles = Vn >> 4;
  if (nTile >= nTiles) return;                  // wave-uniform exit
  const int mTile = blockIdx.y;

  const int m0 = mTile * 16;
  const int n0 = nTile * 16;
  const int row = lane & 15;
  const int kh  = (lane >> 4) << 1;

  const float* __restrict__ aPtr = obj     + (size_t)(m0 + row) * Kn + kh;
  const float* __restrict__ bPtr = emb_ent + (size_t)(n0 + row) * Kn + kh;

  v8f c = {};
  for (int k = 0; k < Kn; k += 4) {
    v2f a = *(const v2f*)(aPtr + k);
    v2f b = *(const v2f*)(bPtr + k);
    c = wmma_f32(a, b, c);
  }

  const int col = n0 + row;
  const float bias = ent_bias[col];
  const int mOff = (lane >> 4) * 8;
  #pragma unroll
  for (int g = 0; g < 8; ++g) {
    float v = c[g] + bias;
    score[(size_t)(m0 + g + mOff) * Vn + col] = 1.0f / (1.0f + expf(-v));
  }
}

// ---------------------------------------------------------------------------
// Launch
// ---------------------------------------------------------------------------
extern "C" void kernel_launch(void* const* d_in, const int* in_sizes, int n_in,
                              void* d_out, int out_size, void* d_ws, size_t ws_size,
                              hipStream_t stream) {
  const int*   edge_src  = (const int*)  d_in[0];
  const int*   edge_dst  = (const int*)  d_in[1];
  const int*   edge_type = (const int*)  d_in[2];
  const int*   head      = (const int*)  d_in[3];
  const int*   rela      = (const int*)  d_in[4];
  const float* edge_norm = (const float*)d_in[5];
  const float* ent_emb   = (const float*)d_in[6];
  const float* rel_emb   = (const float*)d_in[7];
  const float* in_w      = (const float*)d_in[8];
  const float* out_w     = (const float*)d_in[9];
  const float* loop_w    = (const float*)d_in[10];
  const float* w_rel     = (const float*)d_in[11];
  const float* loop_rel  = (const float*)d_in[12];
  const float* conv_bias = (const float*)d_in[13];
  const float* bn_gamma  = (const float*)d_in[14];
  const float* bn_beta   = (const float*)d_in[15];
  const float* emb_ent   = (const float*)d_in[16];
  const float* ent_bias  = (const float*)d_in[17];

  const int E  = in_sizes[0];
  const int Bn = in_sizes[3];
  const int V  = in_sizes[17];
  const int R  = in_sizes[7] / DD;

  // Workspace layout
  float* agg   = (float*)d_ws;                         // [V, OUT]
  float* sum   = agg   + (size_t)V * OUTD;             // [OUT]
  float* sumsq = sum   + OUTD;                         // [OUT]
  float* WTin  = sumsq + OUTD;                         // [NPAD, D]
  float* WTout = WTin  + (size_t)NPAD * DD;            // [NPAD, D]
  float* WTrel = WTout + (size_t)NPAD * DD;            // [NPAD, D]
  float* MT    = WTrel + (size_t)NPAD * DD;            // [NPAD, D]
  float* r_out = MT    + (size_t)NPAD * DD;            // [R, OUT]
  float* obj   = r_out + (size_t)R * OUTD;             // [B, OUT]
  float* score = (float*)d_out;                        // [B, V]

  // 0) build padded/transposed B-operands (+ M_loop circulant fold)
  prep_kernel<<<(NPAD * DD + 255) / 256, 256, 0, stream>>>(
      in_w, out_w, w_rel, loop_rel, loop_w, WTin, WTout, WTrel, MT);

  // 1) self-loop GEMM writes agg (+bias)
  loop_wmma_kernel<<<dim3(V / 16, 2), 256, 0, stream>>>(ent_emb, MT, conv_bias, agg);

  // 2) edge messages: ccorr (LDS) + WMMA projection + atomic scatter
  edge_wmma_kernel<<<E / ET, 256, 0, stream>>>(edge_src, edge_dst, edge_type,
                                               edge_norm, ent_emb, rel_emb,
                                               WTin, WTout, agg, E / 2);

  // 3) BatchNorm (batch stats) + tanh, in place
  hipMemsetAsync(sum, 0, 2 * OUTD * sizeof(float), stream);
  const int nVO = V * OUTD;
  bn_stats_kernel<<<1024, 256, 0, stream>>>(agg, sum, sumsq, nVO);
  bn_apply_kernel<<<(nVO + 255) / 256, 256, 0, stream>>>(agg, sum, sumsq,
                                                         bn_gamma, bn_beta,
                                                         nVO, 1.0f / (float)V);

  // 4) relation projection (WMMA) and query objects
  relout_wmma_kernel<<<dim3(R / 16, 2), 256, 0, stream>>>(rel_emb, WTrel, r_out);
  const int nBO = Bn * OUTD;
  obj_kernel<<<(nBO + 255) / 256, 256, 0, stream>>>(agg, r_out, head, rela, obj, nBO);

  // 5) DistMult decoder GEMM + sigmoid via WMMA
  const int nTiles = V / 16;
  dim3 grid((nTiles + 7) / 8, Bn / 16);
  decoder_wmma_kernel<<<grid, 256, 0, stream>>>(obj, emb_ent, ent_bias, score,
                                                V, OUTD);
}